// GATv2_46377056862923
// MI455X (gfx1250) — compile-verified
//
#include <hip/hip_runtime.h>
#include <hip/hip_bf16.h>
#include <stdint.h>

typedef float v2f __attribute__((ext_vector_type(2)));
typedef float v8f __attribute__((ext_vector_type(8)));

#define DIM 256      // IN_DIM == HEADS*HID == 256 for both layers
#define HID 64
#define HEADS 4
#define OUT_DIM 40
#define NEG_SLOPE 0.2f

// ---------------------------------------------------------------------------
// helpers
// ---------------------------------------------------------------------------
__device__ __forceinline__ float lrelu(float v) { return v > 0.f ? v : NEG_SLOPE * v; }

// order-preserving float<->uint mapping for atomicMax on floats
__device__ __forceinline__ unsigned fenc(float f) {
    unsigned u = __float_as_uint(f);
    return (u & 0x80000000u) ? ~u : (u | 0x80000000u);
}
__device__ __forceinline__ float fdec(unsigned u) {
    return (u & 0x80000000u) ? __uint_as_float(u ^ 0x80000000u) : __uint_as_float(~u);
}

// gfx1250 async global->LDS copy, 16B per lane, per-lane LDS dest address.
// LDS byte address = low 32 bits of the flat pointer to a __shared__ object
// (ISA 10.2: LDS aperture flat addresses truncate to wave-relative LDS addr).
__device__ __forceinline__ void async_g2l_b128(void* lds_dst, const float* gsrc) {
    unsigned ldsoff = (unsigned)(uintptr_t)lds_dst;
    asm volatile("global_load_async_to_lds_b128 %0, %1, off"
                 :: "v"(ldsoff), "v"(gsrc)
                 : "memory");
}
__device__ __forceinline__ void wait_async0() {
    asm volatile("s_wait_asynccnt 0" ::: "memory");
}

// ---------------------------------------------------------------------------
// Dual GEMM: Y{l,r} = X @ W{l,r} + b{l,r}   (X: [N,256], W: [256,256])
// fp32 WMMA 16x16x4. Block = 256 thr (8 waves) -> 16 rows x 128 cols.
// A tile (16x256) and B chunks (32x128) staged in LDS via async-to-LDS.
// grid = (ceil(N/16), 2 /*col half*/, 2 /*l or r*/)
// ---------------------------------------------------------------------------
__global__ __launch_bounds__(256) void gemm_dual(
    const float* __restrict__ X,
    const float* __restrict__ Wl, const float* __restrict__ bl,
    const float* __restrict__ Wr, const float* __restrict__ br,
    float* __restrict__ Yl, float* __restrict__ Yr, int N)
{
    __shared__ float ldsA[16][260];   // 260%64==4 -> A b64 reads hit all 64 banks once
    __shared__ float ldsB[32][136];   // 2*136%64==16 -> half-wave rows in disjoint banks
    const int tid = threadIdx.x;
    const int rowBase = blockIdx.x * 16;
    const int colHalf = blockIdx.y * 128;
    const float* __restrict__ W    = (blockIdx.z == 0) ? Wl : Wr;
    const float* __restrict__ bias = (blockIdx.z == 0) ? bl : br;
    float* __restrict__ Y          = (blockIdx.z == 0) ? Yl : Yr;

    // async stage of 16x256 X tile (float4 per lane, coalesced)
    for (int i = tid; i < 16 * 64; i += 256) {
        int r  = i >> 6;
        int c4 = (i & 63) << 2;
        int rr = rowBase + r; if (rr >= N) rr = N - 1;
        async_g2l_b128(&ldsA[r][c4], X + (size_t)rr * DIM + c4);
    }

    const int wave = tid >> 5;
    const int lane = tid & 31;
    const int g    = lane >> 4;      // half-wave group (K split)
    const int l16  = lane & 15;
    const int col  = colHalf + wave * 16 + l16;

    v8f c = {};
    for (int kc = 0; kc < DIM; kc += 32) {
        // async stage of 32x128 W chunk (32 rows x 32 float4)
        const float* Wb = W + (size_t)kc * DIM + colHalf;
        for (int i = tid; i < 32 * 32; i += 256) {
            int r  = i >> 5;
            int c4 = (i & 31) << 2;
            async_g2l_b128(&ldsB[r][c4], Wb + (size_t)r * DIM + c4);
        }
        wait_async0();
        __syncthreads();

#pragma unroll
        for (int kk = 0; kk < 32; kk += 4) {
            v2f a, b;
            // A fragment: lane m=l16, VGPR v -> K = kc+kk + 2*g + v
            a.x = ldsA[l16][kc + kk + 2 * g + 0];
            a.y = ldsA[l16][kc + kk + 2 * g + 1];
            // B fragment: lane n=l16, VGPR v -> K = kk + 2*g + v
            b.x = ldsB[kk + 2 * g + 0][wave * 16 + l16];
            b.y = ldsB[kk + 2 * g + 1][wave * 16 + l16];
            c = __builtin_amdgcn_wmma_f32_16x16x4_f32(
                    /*neg_a=*/false, a, /*neg_b=*/false, b,
                    /*c_mod=*/(short)0, c, /*reuse_a=*/false, /*reuse_b=*/false);
        }
        __syncthreads();   // protect ldsB before next chunk's async writes
    }

    const float bv = bias[col];
    float* Yp = Y + (size_t)(rowBase + 8 * g) * DIM + col;
    if (rowBase + 16 <= N) {           // full tile fast path (uniform branch)
#pragma unroll
        for (int r = 0; r < 8; ++r) Yp[(size_t)r * DIM] = c[r] + bv;
    } else {
#pragma unroll
        for (int r = 0; r < 8; ++r)
            if (rowBase + r + 8 * g < N) Yp[(size_t)r * DIM] = c[r] + bv;
    }
}

// ---------------------------------------------------------------------------
// per-layer state init: acc = 0, nmax = enc(-inf), denom = 0
// ---------------------------------------------------------------------------
__global__ __launch_bounds__(256) void init_layer(
    float* __restrict__ acc, unsigned* __restrict__ nmax,
    float* __restrict__ ndenom, int accN, int nh)
{
    int i = blockIdx.x * 256 + threadIdx.x;
    if (i < accN) acc[i] = 0.f;
    if (i < nh) { nmax[i] = 0x007FFFFFu; /* enc(-inf) */ ndenom[i] = 0.f; }
}

// ---------------------------------------------------------------------------
// edge logits: e[E,H] = <leaky(xl[src]+xr[dst]), att> ; segment max via atomicMax
// one wave32 per edge: lane covers 8 channels, 8 lanes per head
// ---------------------------------------------------------------------------
__global__ __launch_bounds__(256) void edge_logits(
    const float* __restrict__ xl, const float* __restrict__ xr,
    const int* __restrict__ src, const int* __restrict__ dst,
    const float* __restrict__ att, float* __restrict__ elog,
    unsigned* __restrict__ nmax, int E)
{
    int e = blockIdx.x * 8 + (threadIdx.x >> 5);
    if (e >= E) return;
    int lane = threadIdx.x & 31;
    int s = src[e], d = dst[e];
    int c0 = lane * 8;

    float4 a0 = *(const float4*)(xl + (size_t)s * DIM + c0);
    float4 a1 = *(const float4*)(xl + (size_t)s * DIM + c0 + 4);
    float4 b0 = *(const float4*)(xr + (size_t)d * DIM + c0);
    float4 b1 = *(const float4*)(xr + (size_t)d * DIM + c0 + 4);
    float4 t0 = *(const float4*)(att + c0);
    float4 t1 = *(const float4*)(att + c0 + 4);

    float sum = 0.f;
    sum += lrelu(a0.x + b0.x) * t0.x;
    sum += lrelu(a0.y + b0.y) * t0.y;
    sum += lrelu(a0.z + b0.z) * t0.z;
    sum += lrelu(a0.w + b0.w) * t0.w;
    sum += lrelu(a1.x + b1.x) * t1.x;
    sum += lrelu(a1.y + b1.y) * t1.y;
    sum += lrelu(a1.z + b1.z) * t1.z;
    sum += lrelu(a1.w + b1.w) * t1.w;

    sum += __shfl_xor(sum, 1, 32);
    sum += __shfl_xor(sum, 2, 32);
    sum += __shfl_xor(sum, 4, 32);

    if ((lane & 7) == 0) {
        int h = lane >> 3;
        elog[(size_t)e * HEADS + h] = sum;
        atomicMax(&nmax[(size_t)d * HEADS + h], fenc(sum));
    }
}

// ---------------------------------------------------------------------------
// softmax numerator + segment denom: ex = exp(e - m[dst]); denom += ex
// ---------------------------------------------------------------------------
__global__ __launch_bounds__(256) void edge_softmax(
    float* __restrict__ elog, const int* __restrict__ dst,
    const unsigned* __restrict__ nmax, float* __restrict__ ndenom, int EH)
{
    int i = blockIdx.x * 256 + threadIdx.x;
    if (i >= EH) return;
    int e = i >> 2, h = i & 3;
    int d = dst[e];
    float m = fdec(nmax[(size_t)d * HEADS + h]);
    if (m < -1e30f) m = 0.f;                 // empty-segment guard (matches ref)
    float ex = __expf(elog[i] - m);
    elog[i] = ex;
    atomicAdd(&ndenom[(size_t)d * HEADS + h], ex);
}

// ---------------------------------------------------------------------------
// aggregation: acc[dst] += alpha * xl[src]; one wave per edge, 8 ch/lane
// ---------------------------------------------------------------------------
__global__ __launch_bounds__(256) void edge_aggregate(
    const float* __restrict__ xl, const float* __restrict__ elog,
    const int* __restrict__ src, const int* __restrict__ dst,
    const float* __restrict__ ndenom, float* __restrict__ acc, int E)
{
    int e = blockIdx.x * 8 + (threadIdx.x >> 5);
    if (e >= E) return;
    int lane = threadIdx.x & 31;
    int s = src[e], d = dst[e];
    int h = lane >> 3;
    float ex  = elog[(size_t)e * HEADS + h];
    float den = ndenom[(size_t)d * HEADS + h];
    float alpha = ex / (den + 1e-16f);
    int c0 = lane * 8;
    float4 a0 = *(const float4*)(xl + (size_t)s * DIM + c0);
    float4 a1 = *(const float4*)(xl + (size_t)s * DIM + c0 + 4);
    float* o = acc + (size_t)d * DIM + c0;
    atomicAdd(o + 0, alpha * a0.x);
    atomicAdd(o + 1, alpha * a0.y);
    atomicAdd(o + 2, alpha * a0.z);
    atomicAdd(o + 3, alpha * a0.w);
    atomicAdd(o + 4, alpha * a1.x);
    atomicAdd(o + 5, alpha * a1.y);
    atomicAdd(o + 6, alpha * a1.z);
    atomicAdd(o + 7, alpha * a1.w);
}

// ---------------------------------------------------------------------------
// layer-1 epilogue: h1 = relu(acc + bias1)   (in place)
// ---------------------------------------------------------------------------
__global__ __launch_bounds__(256) void bias_relu(
    float* __restrict__ acc, const float* __restrict__ bias, int total)
{
    int i = blockIdx.x * 256 + threadIdx.x;
    if (i >= total) return;
    acc[i] = fmaxf(acc[i] + bias[i & (DIM - 1)], 0.f);
}

// ---------------------------------------------------------------------------
// layer-2 epilogue: h2 = relu(mean_over_heads(acc) + bias2)   [N,64]
// ---------------------------------------------------------------------------
__global__ __launch_bounds__(256) void head_mean(
    const float* __restrict__ acc, const float* __restrict__ bias2,
    float* __restrict__ h2, int total)
{
    int i = blockIdx.x * 256 + threadIdx.x;
    if (i >= total) return;
    int n = i >> 6, cc = i & 63;
    const float* a = acc + (size_t)n * DIM + cc;
    float s = (a[0] + a[HID] + a[2 * HID] + a[3 * HID]) * 0.25f + bias2[cc];
    h2[i] = fmaxf(s, 0.f);
}

// ---------------------------------------------------------------------------
// classifier: out = h2 @ Wc + bc   [N,40], Wc cached in LDS (64x40)
// ---------------------------------------------------------------------------
__global__ __launch_bounds__(256) void out_final(
    const float* __restrict__ h2, const float* __restrict__ Wc,
    const float* __restrict__ bc, float* __restrict__ out, int N)
{
    __shared__ float sW[HID * OUT_DIM];
    for (int i = threadIdx.x; i < HID * OUT_DIM; i += 256) sW[i] = Wc[i];
    __syncthreads();
    int t = threadIdx.x;
    if (t >= 240) return;                 // 6 nodes x 40 cols per block
    int n = blockIdx.x * 6 + t / OUT_DIM;
    if (n >= N) return;
    int oc = t % OUT_DIM;
    float sum = bc[oc];
    const float* hp = h2 + (size_t)n * HID;
#pragma unroll
    for (int k = 0; k < HID; ++k) sum += hp[k] * sW[k * OUT_DIM + oc];
    out[(size_t)n * OUT_DIM + oc] = sum;
}

// ---------------------------------------------------------------------------
// launch
// ---------------------------------------------------------------------------
extern "C" void kernel_launch(void* const* d_in, const int* in_sizes, int n_in,
                              void* d_out, int out_size, void* d_ws, size_t ws_size,
                              hipStream_t stream) {
    const float* x    = (const float*)d_in[0];
    const int*   src  = (const int*)d_in[1];
    const int*   dst  = (const int*)d_in[2];
    const float* Wl1  = (const float*)d_in[3];
    const float* bl1  = (const float*)d_in[4];
    const float* Wr1  = (const float*)d_in[5];
    const float* br1  = (const float*)d_in[6];
    const float* att1 = (const float*)d_in[7];
    const float* bias1= (const float*)d_in[8];
    const float* Wl2  = (const float*)d_in[9];
    const float* bl2  = (const float*)d_in[10];
    const float* Wr2  = (const float*)d_in[11];
    const float* br2  = (const float*)d_in[12];
    const float* att2 = (const float*)d_in[13];
    const float* bias2= (const float*)d_in[14];
    const float* Wc   = (const float*)d_in[15];
    const float* bc   = (const float*)d_in[16];

    const int N = in_sizes[0] / DIM;
    const int E = in_sizes[1];

    // workspace layout (floats)
    float* ws = (float*)d_ws;
    const size_t A = (size_t)N * DIM;
    float*    xl     = ws;                       // [N,256]
    float*    xr     = ws + A;                   // [N,256]
    float*    acc    = ws + 2 * A;               // [N,256] (h1 after epilogue)
    float*    elog   = ws + 3 * A;               // [E,4]
    unsigned* nmax   = (unsigned*)(elog + (size_t)E * HEADS);  // [N,4]
    float*    ndenom = (float*)(nmax + (size_t)N * HEADS);     // [N,4]
    float*    h2     = ndenom + (size_t)N * HEADS;             // [N,64]
    float*    out    = (float*)d_out;

    const int rowTiles   = (N + 15) / 16;
    const dim3 ggrid(rowTiles, 2, 2);
    const int initBlocks = (int)((A + 255) / 256);
    const int eBlocks    = (E + 7) / 8;
    const int ehBlocks   = (E * HEADS + 255) / 256;

    // ---- layer 1 ----
    gemm_dual<<<ggrid, 256, 0, stream>>>(x, Wl1, bl1, Wr1, br1, xl, xr, N);
    init_layer<<<initBlocks, 256, 0, stream>>>(acc, nmax, ndenom, (int)A, N * HEADS);
    edge_logits<<<eBlocks, 256, 0, stream>>>(xl, xr, src, dst, att1, elog, nmax, E);
    edge_softmax<<<ehBlocks, 256, 0, stream>>>(elog, dst, nmax, ndenom, E * HEADS);
    edge_aggregate<<<eBlocks, 256, 0, stream>>>(xl, elog, src, dst, ndenom, acc, E);
    bias_relu<<<initBlocks, 256, 0, stream>>>(acc, bias1, (int)A);

    // ---- layer 2 ----
    gemm_dual<<<ggrid, 256, 0, stream>>>(acc, Wl2, bl2, Wr2, br2, xl, xr, N);
    init_layer<<<initBlocks, 256, 0, stream>>>(acc, nmax, ndenom, (int)A, N * HEADS);
    edge_logits<<<eBlocks, 256, 0, stream>>>(xl, xr, src, dst, att2, elog, nmax, E);
    edge_softmax<<<ehBlocks, 256, 0, stream>>>(elog, dst, nmax, ndenom, E * HEADS);
    edge_aggregate<<<eBlocks, 256, 0, stream>>>(xl, elog, src, dst, ndenom, acc, E);
    head_mean<<<(N * HID + 255) / 256, 256, 0, stream>>>(acc, bias2, h2, N * HID);

    // ---- classifier ----
    out_final<<<(N + 5) / 6, 256, 0, stream>>>(h2, Wc, bc, out, N);
}